// LIFNode_14242111553593
// MI455X (gfx1250) — compile-verified
//
#include <hip/hip_runtime.h>
#include <stdint.h>

// LIF forward scan, [B,C,N,T] f32, T=32 contiguous innermost.
// decay = 1 - 1/TAU = 0.5, VTH = 1.0.
// Memory-bound streaming op: 512 MB total traffic -> ~22us @ 23.3 TB/s.
// CDNA5 path: async global->LDS DMA (ASYNCcnt) with padded-scatter LDS layout,
// per-wave double buffering (no barriers), nontemporal vector stores.

typedef float v4f __attribute__((ext_vector_type(4)));
typedef int   v4i __attribute__((ext_vector_type(4)));

// Address-space-qualified v4i for the async-copy builtin:
// clang prints AS1 as "__device__", AS3 as "__shared__".
typedef __attribute__((address_space(1))) v4i gv4i;
typedef __attribute__((address_space(3))) v4i lv4i;

#define WAVES_PER_BLOCK 8
#define BLOCK_THREADS   (WAVES_PER_BLOCK * 32)
#define T_LEN           32
#define GROUP_NEURONS   32                           // neurons per wave-group
#define GROUP_ELEMS     (GROUP_NEURONS * T_LEN)      // 1024 floats = 4KB
#define ROW_F           36                           // 32 data + 4 pad floats (144B row)
#define BUF_F           (GROUP_NEURONS * ROW_F)      // 1152 floats / buffer
#define WAVE_LDS_F      (2 * BUF_F)                  // double buffered
#define BLOCK_LDS_F     (WAVES_PER_BLOCK * WAVE_LDS_F) // 18432 floats = 73728 B

__device__ __forceinline__ void async_copy_b128(const float* g, float* l) {
#if __has_builtin(__builtin_amdgcn_global_load_async_to_lds_b128)
  __builtin_amdgcn_global_load_async_to_lds_b128((gv4i*)g, (lv4i*)l, 0, 0);
#else
  __attribute__((address_space(3))) char* l3 =
      (__attribute__((address_space(3))) char*)l;
  unsigned int loff = (unsigned int)(unsigned long long)l3;
  asm volatile("global_load_async_to_lds_b128 %0, %1, off"
               :: "v"(loff), "v"(g) : "memory");
#endif
}

__device__ __forceinline__ void wait_async8() {
#if __has_builtin(__builtin_amdgcn_s_wait_asynccnt)
  __builtin_amdgcn_s_wait_asynccnt(8);
#else
  asm volatile("s_wait_asynccnt 0x8" ::: "memory");
#endif
}

__device__ __forceinline__ void wait_async0() {
#if __has_builtin(__builtin_amdgcn_s_wait_asynccnt)
  __builtin_amdgcn_s_wait_asynccnt(0);
#else
  asm volatile("s_wait_asynccnt 0x0" ::: "memory");
#endif
}

__global__ __launch_bounds__(BLOCK_THREADS)
void lif_fwd_kernel(const float* __restrict__ x, float* __restrict__ out,
                    int numGroups) {
  __shared__ float lds[BLOCK_LDS_F];

  const int lane = threadIdx.x & 31;
  const int wave = threadIdx.x >> 5;
  float* wlds = &lds[wave * WAVE_LDS_F];

  const int waveId  = blockIdx.x * WAVES_PER_BLOCK + wave;
  const int waveCnt = gridDim.x * WAVES_PER_BLOCK;

  // Async-copy scatter map: issue k moves 512B coalesced; lane supplies the
  // padded LDS destination.  Global elem offset = k*128 + lane*4 ->
  // neuron row = k*4 + lane/8, intra-row elem = (lane%8)*4.
  const int nsub = lane >> 3;
  const int isub = (lane & 7) * 4;

  int g = waveId;
  int buf = 0;

  if (g < numGroups) {
    const float* gb = x + (size_t)g * GROUP_ELEMS;
    float* lb = wlds;  // buffer 0
#pragma unroll
    for (int k = 0; k < 8; ++k)
      async_copy_b128(gb + k * 128 + lane * 4,
                      lb + (k * 4 + nsub) * ROW_F + isub);
  }

  for (; g < numGroups; g += waveCnt) {
    const int gn   = g + waveCnt;
    const int nbuf = buf ^ 1;

    if (gn < numGroups) {
      const float* gb = x + (size_t)gn * GROUP_ELEMS;
      float* lb = wlds + nbuf * BUF_F;
#pragma unroll
      for (int k = 0; k < 8; ++k)
        async_copy_b128(gb + k * 128 + lane * 4,
                        lb + (k * 4 + nsub) * ROW_F + isub);
      // Async loads retire in order (per-wave counter): waiting <=8 means the
      // current buffer's 8 copies are done; the next group's 8 stay in flight.
      wait_async8();
    } else {
      wait_async0();
    }
    asm volatile("" ::: "memory");

    // Serial LIF scan: lane owns neuron (g*32 + lane); its 32 timesteps sit in
    // one padded LDS row (144B stride -> optimal 2-way b128 LDS access).
    const float* row = wlds + buf * BUF_F + lane * ROW_F;
    float* ob = out + (size_t)g * GROUP_ELEMS + lane * T_LEN;

    float v = 0.0f;
#pragma unroll
    for (int j = 0; j < 8; ++j) {
      v4f xv = *(const v4f*)(row + j * 4);
      v4f s;
#pragma unroll
      for (int c = 0; c < 4; ++c) {
        v = v * 0.5f + xv[c];                 // leaky integrate (decay = 0.5)
        const float sp = (v > 1.0f) ? 1.0f : 0.0f;  // fire at VTH = 1
        v -= sp;                              // soft reset (s * VTH)
        s[c] = sp;
      }
      __builtin_nontemporal_store(s, (v4f*)(ob + j * 4));  // streaming store
    }
    asm volatile("" ::: "memory");
    buf = nbuf;
  }
}

extern "C" void kernel_launch(void* const* d_in, const int* in_sizes, int n_in,
                              void* d_out, int out_size, void* d_ws,
                              size_t ws_size, hipStream_t stream) {
  (void)n_in; (void)d_ws; (void)ws_size; (void)out_size;
  const float* x = (const float*)d_in[0];
  float* out = (float*)d_out;

  const long long total = (long long)in_sizes[0];      // B*C*N*T elements
  const int numGroups = (int)(total / GROUP_ELEMS);    // 65536 for ref shape

  int blocks = (numGroups + WAVES_PER_BLOCK - 1) / WAVES_PER_BLOCK;
  if (blocks > 2048) blocks = 2048;   // persistent waves, ~4 groups each
  if (blocks < 1) blocks = 1;

  lif_fwd_kernel<<<blocks, BLOCK_THREADS, 0, stream>>>(x, out, numGroups);
}